// DenseBlockEnd_13408887898713
// MI455X (gfx1250) — compile-verified
//
#include <hip/hip_runtime.h>
#include <cstdint>

// out = relu( (a < n_atoms[b]) ? features[b,a,:] + res[0,b,a,:] + res[1,b,a,:] : 0 )
// Shapes: features (256,128,1024) f32; residuals (2,256,128,1024) f32;
//         mol_slice (256,2) int (col 0 = n_atoms). Output (256,128,1024) f32.
//
// Memory-bound streaming kernel. One block == one (b,a) row of 1024 floats:
// 256 threads x float4. Mask is uniform per row -> masked rows skip ALL input
// reads and store zeros (cuts expected read traffic roughly in half).
// Residual streams are staged LDS-side via CDNA5 async DMA
// (global_load_async_to_lds_b128, ASYNCcnt) with non-temporal hints; the
// features stream uses a direct non-temporal b128 load overlapped with the
// async DMAs.

typedef __attribute__((ext_vector_type(4))) float v4f;

static constexpr int    kBatch    = 256;
static constexpr int    kMaxAtom  = 128;
static constexpr int    kFeat     = 1024;
static constexpr int    kRowF4    = kFeat / 4;                  // 256 float4 per row
static constexpr int    kRows     = kBatch * kMaxAtom;          // 32768 rows
static constexpr size_t kStreamF4 = (size_t)kRows * kRowF4;     // float4 per (B,A,F) tensor

__global__ __launch_bounds__(256)
void dense_block_end_kernel(const v4f* __restrict__ feat,
                            const v4f* __restrict__ res,
                            const int* __restrict__ mol,
                            v4f* __restrict__ out)
{
    __shared__ v4f s_r0[256];
    __shared__ v4f s_r1[256];

    const uint32_t row  = blockIdx.x;            // flat (b, a) row
    const uint32_t b    = row >> 7;              // / 128
    const uint32_t a    = row & (kMaxAtom - 1);  // % 128
    const uint32_t t    = threadIdx.x;           // 0..255
    const size_t   elem = (size_t)row * kRowF4 + t;

    // mol_slice[b][0] = n_atoms[b]; uniform per block -> scalar load
    const int m = mol[2 * b];

    if ((int)a >= m) {
        // Masked row: write zeros, skip all input reads.
        v4f z = (v4f)0.0f;
        __builtin_nontemporal_store(z, &out[elem]);
        return;
    }

    // ---- CDNA5 async DMA: residual streams -> LDS (tracked by ASYNCcnt) ----
    const v4f* r0p = res + elem;               // residuals[0, b, a, 4t..4t+3]
    const v4f* r1p = res + kStreamF4 + elem;   // residuals[1, b, a, 4t..4t+3]
    uint32_t lds0 = (uint32_t)(uintptr_t)&s_r0[t];  // low 32 bits == LDS byte offset
    uint32_t lds1 = (uint32_t)(uintptr_t)&s_r1[t];

    asm volatile("global_load_async_to_lds_b128 %0, %1, off th:TH_LOAD_NT"
                 :: "v"(lds0), "v"(r0p) : "memory");
    asm volatile("global_load_async_to_lds_b128 %0, %1, off th:TH_LOAD_NT"
                 :: "v"(lds1), "v"(r1p) : "memory");

    // Direct non-temporal load of the features stream, overlapped with the DMAs.
    v4f f = __builtin_nontemporal_load(&feat[elem]);

    // Wait for this wave's async DMAs; each lane reads back its own LDS slot,
    // so no workgroup barrier is required.
    asm volatile("s_wait_asynccnt 0" ::: "memory");

    v4f r0 = s_r0[t];
    v4f r1 = s_r1[t];

    v4f s = f + r0 + r1;
    v4f o;
    o.x = fmaxf(s.x, 0.0f);
    o.y = fmaxf(s.y, 0.0f);
    o.z = fmaxf(s.z, 0.0f);
    o.w = fmaxf(s.w, 0.0f);

    __builtin_nontemporal_store(o, &out[elem]);
}

extern "C" void kernel_launch(void* const* d_in, const int* in_sizes, int n_in,
                              void* d_out, int out_size, void* d_ws, size_t ws_size,
                              hipStream_t stream) {
    (void)in_sizes; (void)n_in; (void)out_size; (void)d_ws; (void)ws_size;

    const v4f* feat = (const v4f*)d_in[0];   // features  (256,128,1024) f32
    const v4f* res  = (const v4f*)d_in[1];   // residuals (2,256,128,1024) f32
    const int* mol  = (const int*)d_in[2];   // mol_slice (256,2) int
    v4f*       out  = (v4f*)d_out;           // (256,128,1024) f32

    dense_block_end_kernel<<<dim3(kRows), dim3(256), 0, stream>>>(feat, res, mol, out);
}